// CalibrationError_5068061409627
// MI455X (gfx1250) — compile-verified
//
#include <hip/hip_runtime.h>
#include <hip/hip_bf16.h>
#include <math.h>

// Calibration error (ECE/MCE) for [N,100] f32 logits.
// Bandwidth-bound: 800MB stream ~34us @ 23.3TB/s. Data path uses CDNA5
// async global->LDS copies (ASYNCcnt) with double buffering; per-lane row
// reduction from LDS; deterministic integer histogram accumulation.

#define NBINS      15
#define NCLS       100
#define ROW_BYTES  (NCLS * 4)            // 400
#define TILE_ROWS  32
#define TILE_BYTES (TILE_ROWS * ROW_BYTES) // 12800 = 25 * 512
#define CHUNKS     (TILE_BYTES / 512)    // 25 wave-wide b128 async loads
#define WAVES_PB   4
#define TPB        (WAVES_PB * 32)
#define NBLOCKS    2048
#define HIST_SLOTS (3 * NBINS)           // cnt[15], confQ[15], acc[15]

// Issue one 32-row tile (12800 contiguous bytes) as 25 async b128 copies.
// INST_OFFSET applies to both the global and LDS address (ISA 08 §4.4),
// so a single VGPR address pair + immediate offsets covers the whole tile.
__device__ __forceinline__ void async_copy_tile(unsigned long long gbase,
                                                unsigned int ldsbase,
                                                int lane) {
  unsigned long long g = gbase + (unsigned int)(lane * 16);
  unsigned int       l = ldsbase + (unsigned int)(lane * 16);
#pragma unroll
  for (int i = 0; i < CHUNKS; ++i) {
    asm volatile("global_load_async_to_lds_b128 %0, %1, off offset:%2"
                 :: "v"(l), "v"(g), "i"(i * 512)
                 : "memory");
  }
}

__device__ __forceinline__ void wait_async_le_tile() {
  // one full tile (25 loads) may remain outstanding; loads complete in order
  asm volatile("s_wait_asynccnt 25" ::: "memory");
}
__device__ __forceinline__ void wait_async_zero() {
  asm volatile("s_wait_asynccnt 0" ::: "memory");
}

// One lane fully owns one row (100 floats = 25 float4).
__device__ __forceinline__ void process_row(const float4* __restrict__ rowp,
                                            int label,
                                            unsigned long long* __restrict__ hist) {
  float4 c[CHUNKS];
#pragma unroll
  for (int j = 0; j < CHUNKS; ++j) c[j] = rowp[j];

  // max + first-occurrence argmax (strict > keeps earliest index)
  float m = c[0].x;
  int arg = 0;
#pragma unroll
  for (int j = 0; j < CHUNKS; ++j) {
    float4 v = c[j];
    if (v.x > m) { m = v.x; arg = 4 * j + 0; }
    if (v.y > m) { m = v.y; arg = 4 * j + 1; }
    if (v.z > m) { m = v.z; arg = 4 * j + 2; }
    if (v.w > m) { m = v.w; arg = 4 * j + 3; }
  }

  // conf = max(softmax) = 1 / sum(exp(x - max));  s >= 1 so conf <= 1
  float s = 0.0f;
#pragma unroll
  for (int j = 0; j < CHUNKS; ++j) {
    float4 v = c[j];
    s += __expf(v.x - m) + __expf(v.y - m) + __expf(v.z - m) + __expf(v.w - m);
  }
  float conf = 1.0f / s;

  // bin = clip(ceil(conf*15) - 1, 0, 14)  — matches reference
  int bin = (int)ceilf(conf * (float)NBINS) - 1;
  bin = bin < 0 ? 0 : (bin > NBINS - 1 ? NBINS - 1 : bin);

  // deterministic integer accumulation; conf in 32.32 fixed point
  unsigned long long q = (unsigned long long)((double)conf * 4294967296.0);
  atomicAdd(&hist[bin], 1ull);                                   // count
  atomicAdd(&hist[NBINS + bin], q);                              // sum conf (Q32)
  atomicAdd(&hist[2 * NBINS + bin],
            (unsigned long long)(arg == label ? 1 : 0));         // sum acc
}

__global__ void calib_init_kernel(unsigned long long* __restrict__ ws) {
  if (threadIdx.x < HIST_SLOTS) ws[threadIdx.x] = 0ull;
}

__global__ void __launch_bounds__(TPB)
calib_main_kernel(const float* __restrict__ logits,
                  const int* __restrict__ labels,
                  unsigned long long* __restrict__ ws, int N) {
  __shared__ __align__(16) unsigned char smem[2][WAVES_PB][TILE_BYTES];
  __shared__ unsigned long long hist[HIST_SLOTS];
  if (threadIdx.x < HIST_SLOTS) hist[threadIdx.x] = 0ull;
  __syncthreads();

  const int lane   = threadIdx.x & 31;
  const int wib    = threadIdx.x >> 5;
  const int gwave  = blockIdx.x * WAVES_PB + wib;
  const int nwaves = gridDim.x * WAVES_PB;
  const int ntiles = N / TILE_ROWS;
  const int rem    = N - ntiles * TILE_ROWS;

  const unsigned long long gbase = (unsigned long long)(uintptr_t)logits;
  // low 32 bits of the generic LDS pointer == byte offset in wave's LDS
  const unsigned int lds0 = (unsigned int)(uintptr_t)(&smem[0][wib][0]);
  const unsigned int lds1 = (unsigned int)(uintptr_t)(&smem[1][wib][0]);

  int tile = gwave;
  int p = 0;
  if (tile < ntiles)
    async_copy_tile(gbase + (unsigned long long)tile * TILE_BYTES, lds0, lane);

  for (; tile < ntiles; tile += nwaves) {
    const int nt = tile + nwaves;
    if (nt < ntiles) {
      async_copy_tile(gbase + (unsigned long long)nt * TILE_BYTES,
                      p ? lds0 : lds1, lane);
      wait_async_le_tile();   // current tile resident; next tile in flight
    } else {
      wait_async_zero();
    }
    const unsigned char* buf = &smem[p][wib][0];
    const float4* rowp = (const float4*)(buf + lane * ROW_BYTES);
    const int row = tile * TILE_ROWS + lane;
    process_row(rowp, labels[row], hist);
    p ^= 1;
  }

  // tail rows (N % 32): direct global loads, negligible volume
  const int tid = blockIdx.x * blockDim.x + threadIdx.x;
  if (tid < rem) {
    const int row = ntiles * TILE_ROWS + tid;
    const float4* rowp = (const float4*)(logits + (size_t)row * NCLS);
    process_row(rowp, labels[row], hist);
  }

  __syncthreads();
  if (threadIdx.x < HIST_SLOTS)
    atomicAdd(&ws[threadIdx.x], hist[threadIdx.x]);
}

__global__ void calib_final_kernel(const unsigned long long* __restrict__ ws,
                                   float* __restrict__ out, double n_total) {
  if (threadIdx.x == 0 && blockIdx.x == 0) {
    double ece = 0.0, mce = -1.0;
    bool any = false;
    for (int b = 0; b < NBINS; ++b) {
      double cnt = (double)ws[b];
      if (cnt > 0.0) {
        double sc  = (double)ws[NBINS + b] * (1.0 / 4294967296.0);
        double sa  = (double)ws[2 * NBINS + b];
        double gap = fabs(sc / cnt - sa / cnt);
        ece += gap * cnt / n_total;
        if (gap > mce) mce = gap;
        any = true;
      }
    }
    out[0] = (float)ece;
    out[1] = any ? (float)mce : 1.0f;
  }
}

extern "C" void kernel_launch(void* const* d_in, const int* in_sizes, int n_in,
                              void* d_out, int out_size, void* d_ws, size_t ws_size,
                              hipStream_t stream) {
  const float* logits = (const float*)d_in[0];
  const int*   labels = (const int*)d_in[1];
  const int N = in_sizes[1];                 // labels count == rows
  unsigned long long* ws = (unsigned long long*)d_ws;  // 45 * u64 = 360 B
  float* out = (float*)d_out;                // [ece, mce]

  calib_init_kernel<<<1, 64, 0, stream>>>(ws);
  calib_main_kernel<<<NBLOCKS, TPB, 0, stream>>>(logits, labels, ws, N);
  calib_final_kernel<<<1, 32, 0, stream>>>(ws, out, (double)N);
}